// SO3UFormer_66305705116316
// MI455X (gfx1250) — compile-verified
//
#include <hip/hip_runtime.h>
#include <cstdint>
#include <cstddef>

// Problem constants (from reference)
#define B_     2
#define N_IN   163842
#define N_OUT  40962
#define C_     128
#define K_     7
// -1 / (2 * 0.4^2)
#define GAUSS_COEF (-3.125f)

// Native non-returning fp32 atomic add (L2 atomic unit), guaranteed lowering.
__device__ __forceinline__ void atomic_add_f32_native(float* p, float v) {
    asm volatile("global_atomic_add_f32 %0, %1, off"
                 :: "v"((uint64_t)(uintptr_t)p), "v"(v)
                 : "memory");
}

// ---------------------------------------------------------------------------
// Kernel 1: zero workspace (down + denom). b128 stores + scalar tail.
// ---------------------------------------------------------------------------
__global__ void k_zero(float* __restrict__ ws, long long n) {
    long long i = (long long)blockIdx.x * blockDim.x + threadIdx.x;
    const long long n4 = n >> 2;                 // # of full float4 chunks
    if (i < n4) {
        ((float4*)ws)[i] = make_float4(0.f, 0.f, 0.f, 0.f);
    } else {
        long long t = (n4 << 2) + (i - n4);      // tail elements
        if (t < n) ws[t] = 0.0f;
    }
}

// ---------------------------------------------------------------------------
// Kernel 2: scatter.  One 128-thread block per input node n; thread c handles
// channel c for both batches. Native fp32 atomics land in L2 (down = 42MB,
// L2-resident; avg 4-way collisions resolved by the L2 atomic units).
// ---------------------------------------------------------------------------
__global__ __launch_bounds__(C_) void k_scatter(const float* __restrict__ x,
                                                const float* __restrict__ omega,
                                                const int*   __restrict__ parent,
                                                float* __restrict__ down,
                                                float* __restrict__ denom) {
    const int n = blockIdx.x;
    const int c = threadIdx.x;
    const float om = omega[n];
    const int   p  = parent[n];

    const float v0 = x[(size_t)0 * N_IN * C_ + (size_t)n * C_ + c] * om;
    const float v1 = x[(size_t)1 * N_IN * C_ + (size_t)n * C_ + c] * om;

    atomic_add_f32_native(&down[(size_t)0 * N_OUT * C_ + (size_t)p * C_ + c], v0);
    atomic_add_f32_native(&down[(size_t)1 * N_OUT * C_ + (size_t)p * C_ + c], v1);
    if (c == 0) {
        atomic_add_f32_native(&denom[p], om);
    }
}

// ---------------------------------------------------------------------------
// Kernel 3: normalize down by clip(denom, 1e-8). float4 per thread.
// ---------------------------------------------------------------------------
__global__ void k_norm(float* __restrict__ down, const float* __restrict__ denom) {
    const long long total4 = (long long)B_ * N_OUT * C_ / 4;
    long long i = (long long)blockIdx.x * blockDim.x + threadIdx.x;
    if (i >= total4) return;
    const long long e = i * 4;                         // element index
    const int p = (int)((e / C_) % N_OUT);             // output-node id
    const float inv = 1.0f / fmaxf(denom[p], 1e-8f);
    float4* d4 = (float4*)down;
    float4 v = d4[i];
    v.x *= inv; v.y *= inv; v.z *= inv; v.w *= inv;
    d4[i] = v;
}

// ---------------------------------------------------------------------------
// Kernel 4: gather + weighted sum.  One wave32 per input node n.
//   - Load cand_idx / delta / cand_mask first (VMEM in flight),
//   - issue 14 async global->LDS b128 row copies (ASYNCcnt path, CDNA5),
//     each instruction moves one 512B row of `down` (lane -> 16B),
//   - overlap: Gaussian weights (v_exp_f32) + normalization,
//   - single s_wait_asynccnt 0, then float4 FMA reduction from LDS.
// ---------------------------------------------------------------------------
__global__ __launch_bounds__(32) void k_gather(const float* __restrict__ delta,
                                               const float* __restrict__ cmask,
                                               const int*   __restrict__ cidx,
                                               const float* __restrict__ down,
                                               float* __restrict__ out) {
    __shared__ __align__(16) float lds[B_ * K_ * C_];   // 2*7*128 floats = 7 KB

    const int n    = blockIdx.x;
    const int lane = threadIdx.x;                       // 0..31, wave32

    // LDS byte offset of the staging buffer (generic ptr low 32 bits == LDS addr)
    const uint32_t ldsBase = (uint32_t)(uintptr_t)(&lds[0]);

    // Kick off all scalar-ish metadata loads first so they are in flight.
    int   idx[K_];
    float dl[K_], mk[K_];
#pragma unroll
    for (int k = 0; k < K_; ++k) idx[k] = cidx[n * K_ + k];
#pragma unroll
    for (int k = 0; k < K_; ++k) { dl[k] = delta[n * K_ + k]; mk[k] = cmask[n * K_ + k]; }

    // Fire all 14 async row copies: per-lane 16B, 512B per instruction.
#pragma unroll
    for (int b = 0; b < B_; ++b) {
#pragma unroll
        for (int k = 0; k < K_; ++k) {
            const float* src = down + ((size_t)b * N_OUT + (size_t)idx[k]) * C_ + lane * 4;
            const uint32_t dst = ldsBase + (uint32_t)(((b * K_ + k) * C_ + lane * 4) * sizeof(float));
            asm volatile("global_load_async_to_lds_b128 %0, %1, off"
                         :: "v"(dst), "v"((uint64_t)(uintptr_t)src)
                         : "memory");
        }
    }

    // Overlapped VALU work: Gaussian candidate weights, normalized over K.
    float w[K_];
    float s = 0.0f;
#pragma unroll
    for (int k = 0; k < K_; ++k) {
        const float e = __expf(dl[k] * dl[k] * GAUSS_COEF) * mk[k];
        w[k] = e;
        s += e;
    }
    const float invs = 1.0f / fmaxf(s, 1e-8f);
#pragma unroll
    for (int k = 0; k < K_; ++k) w[k] *= invs;

    // Wait for the async engine to finish writing LDS.
    asm volatile("s_wait_asynccnt 0x0" ::: "memory");

    // Weighted reduction over K from LDS, float4 per lane, both batches.
#pragma unroll
    for (int b = 0; b < B_; ++b) {
        float4 acc = make_float4(0.f, 0.f, 0.f, 0.f);
#pragma unroll
        for (int k = 0; k < K_; ++k) {
            const float4 g = *(const float4*)&lds[(b * K_ + k) * C_ + lane * 4];
            acc.x = fmaf(w[k], g.x, acc.x);
            acc.y = fmaf(w[k], g.y, acc.y);
            acc.z = fmaf(w[k], g.z, acc.z);
            acc.w = fmaf(w[k], g.w, acc.w);
        }
        *(float4*)&out[((size_t)b * N_IN + n) * C_ + lane * 4] = acc;
    }
}

// ---------------------------------------------------------------------------
// Launcher
// ---------------------------------------------------------------------------
extern "C" void kernel_launch(void* const* d_in, const int* in_sizes, int n_in,
                              void* d_out, int out_size, void* d_ws, size_t ws_size,
                              hipStream_t stream) {
    const float* x      = (const float*)d_in[0];   // (B, N_IN, C) f32
    const float* omega  = (const float*)d_in[1];   // (N_IN,)      f32
    const float* delta  = (const float*)d_in[2];   // (N_IN, K)    f32
    const float* cmask  = (const float*)d_in[3];   // (N_IN, K)    f32
    const int*   parent = (const int*)d_in[4];     // (N_IN,)      i32
    const int*   cidx   = (const int*)d_in[5];     // (N_IN, K)    i32
    float*       out    = (float*)d_out;           // (B, N_IN, C) f32

    float* down  = (float*)d_ws;                               // B*N_OUT*C f32
    float* denom = down + (size_t)B_ * N_OUT * C_;             // N_OUT f32

    const long long zn      = (long long)B_ * N_OUT * C_ + N_OUT;
    const long long zthreads = (zn >> 2) + (zn & 3);           // float4 chunks + tail
    k_zero<<<(unsigned)((zthreads + 255) / 256), 256, 0, stream>>>((float*)d_ws, zn);

    k_scatter<<<N_IN, C_, 0, stream>>>(x, omega, parent, down, denom);

    const long long nq = (long long)B_ * N_OUT * C_ / 4;
    k_norm<<<(unsigned)((nq + 255) / 256), 256, 0, stream>>>(down, denom);

    k_gather<<<N_IN, 32, 0, stream>>>(delta, cmask, cidx, down, out);
}